// MH_GAT_21345987461372
// MI455X (gfx1250) — compile-verified
//
#include <hip/hip_runtime.h>
#include <math.h>

// ---------------- problem constants (match reference) ----------------
#define N_NODES   4096
#define E_EDGES   86016          // 4096*20 + 4096 self loops; exact multiple of 32
#define HEADS     7
#define CH        300
#define D_IN      256
#define D_DEG     8
#define D_GAT     272            // 256 + 2*8, multiple of 4
#define D_HID     2100           // 7*300
#define D_HID_PAD 2112           // next multiple of 16 (132 tiles, even)
#define D_OUT     256
#define NEG_SLOPE 0.2f
#define BN_EPS    1e-5
#define MAXDEG    512            // in-degree bound (mean ~21, Binomial tail -> safe)

typedef float v2f __attribute__((ext_vector_type(2)));
typedef float v8f __attribute__((ext_vector_type(8)));

// ---------------- degree counting (integer atomics: deterministic) ----------------
__global__ void zero_deg_kernel(int* __restrict__ in_deg, int* __restrict__ out_deg) {
    int i = blockIdx.x * blockDim.x + threadIdx.x;
    if (i < N_NODES) { in_deg[i] = 0; out_deg[i] = 0; }
}

__global__ void count_deg_kernel(const int* __restrict__ src, const int* __restrict__ dst,
                                 int* __restrict__ in_deg, int* __restrict__ out_deg) {
    int e = blockIdx.x * blockDim.x + threadIdx.x;
    if (e < E_EDGES) {
        atomicAdd(&out_deg[src[e]], 1);
        atomicAdd(&in_deg[dst[e]], 1);
    }
}

// ---------------- xc = [x | emb_in[in_deg] | emb_out[out_deg]]  (N x 272) ----------------
__global__ void build_xc_kernel(const float* __restrict__ x,
                                const float* __restrict__ emb_in,
                                const float* __restrict__ emb_out,
                                const int* __restrict__ in_deg,
                                const int* __restrict__ out_deg,
                                float* __restrict__ xc) {
    int n = blockIdx.x;
    int t = threadIdx.x;                    // blockDim.x == 272
    float v;
    if (t < D_IN) {
        v = x[(size_t)n * D_IN + t];
    } else if (t < D_IN + D_DEG) {
        int d = in_deg[n]; d = d < (N_NODES - 1) ? d : (N_NODES - 1);
        v = emb_in[(size_t)d * D_DEG + (t - D_IN)];
    } else {
        int d = out_deg[n]; d = d < (N_NODES - 1) ? d : (N_NODES - 1);
        v = emb_out[(size_t)d * D_DEG + (t - D_IN - D_DEG)];
    }
    xc[(size_t)n * D_GAT + t] = v;
}

// ---------------- weight transposes (make B fragments contiguous float2) ----------------
// WgT[c][k] = W_gat[k][c], c in [0,2112) (rows >= 2100 zero-padded), k in [0,272)
__global__ void transpose_wgat_kernel(const float* __restrict__ W, float* __restrict__ WT) {
    int i = blockIdx.x * blockDim.x + threadIdx.x;
    if (i < D_HID_PAD * D_GAT) {
        int c = i / D_GAT, k = i % D_GAT;
        WT[i] = (c < D_HID) ? W[(size_t)k * D_HID + c] : 0.0f;
    }
}
// W3T[c][k] = W3[k][c], c in [0,256), k in [0,2100)
__global__ void transpose_w3_kernel(const float* __restrict__ W, float* __restrict__ WT) {
    int i = blockIdx.x * blockDim.x + threadIdx.x;
    if (i < D_OUT * D_HID) {
        int c = i / D_HID, k = i % D_HID;
        WT[i] = W[(size_t)k * D_OUT + c];
    }
}

// ---------------- fp32 WMMA GEMM, 2x2 register-blocked ----------------
// Each wave computes a 32x32 output block (4 accumulators). Per K-step:
// 4 float2 loads feed 4 V_WMMA_F32_16X16X4_F32 ops (A/B fragments each reused
// twice -> 8 FLOP/byte vs 4 for the unblocked version). A row-major (lda),
// BT row-major with rows = output columns (ldb), D row-major (ldd).
// All branches are wave-uniform -> EXEC all ones inside WMMA (ISA 7.12).
// Requires: M % 32 == 0, nTilesN even.
__global__ void gemm_wmma_f32_2x2_kernel(const float* __restrict__ A,
                                         const float* __restrict__ BT,
                                         float* __restrict__ D,
                                         const float* __restrict__ bias, // nullptr -> none
                                         int M, int K, int nTilesN,
                                         int lda, int ldb, int ldd, int relu) {
    const int lane = threadIdx.x & 31;
    const int wv   = threadIdx.x >> 5;
    const int nPairsN = nTilesN >> 1;
    const int nPairs  = (M >> 5) * nPairsN;
    const int pair = blockIdx.x * (blockDim.x >> 5) + wv;
    if (pair >= nPairs) return;                 // uniform per wave
    const int tm2 = pair / nPairsN;             // 32-row block index
    const int tn2 = pair % nPairsN;             // 32-col block index
    const int hi = lane >> 4;                   // lane half selects K pair
    const int lo = lane & 15;

    const float* arow0 = A  + (size_t)(tm2 * 32 + lo) * lda + 2 * hi;
    const float* arow1 = arow0 + (size_t)16 * lda;
    const float* brow0 = BT + (size_t)(tn2 * 32 + lo) * ldb + 2 * hi;
    const float* brow1 = brow0 + (size_t)16 * ldb;

    v8f acc00 = {}, acc01 = {}, acc10 = {}, acc11 = {};
    for (int k = 0; k < K; k += 4) {
        v2f a0 = *(const v2f*)(arow0 + k);      // 8B-aligned: lda, k, 2*hi all even
        v2f a1 = *(const v2f*)(arow1 + k);
        v2f b0 = *(const v2f*)(brow0 + k);
        v2f b1 = *(const v2f*)(brow1 + k);
        acc00 = __builtin_amdgcn_wmma_f32_16x16x4_f32(false, a0, false, b0, (short)0, acc00, false, false);
        acc01 = __builtin_amdgcn_wmma_f32_16x16x4_f32(false, a0, false, b1, (short)0, acc01, false, false);
        acc10 = __builtin_amdgcn_wmma_f32_16x16x4_f32(false, a1, false, b0, (short)0, acc10, false, false);
        acc11 = __builtin_amdgcn_wmma_f32_16x16x4_f32(false, a1, false, b1, (short)0, acc11, false, false);
    }

    // epilogue: C/D layout per sub-tile: M = v + 8*hi, N = lane&15
    const float bv0 = bias ? bias[tn2 * 32 + lo]      : 0.0f;
    const float bv1 = bias ? bias[tn2 * 32 + 16 + lo] : 0.0f;
    #pragma unroll
    for (int r = 0; r < 2; ++r) {
        const v8f* accs0 = (r == 0) ? &acc00 : &acc10;
        const v8f* accs1 = (r == 0) ? &acc01 : &acc11;
        float* drow = D + (size_t)(tm2 * 32 + r * 16 + 8 * hi) * ldd + tn2 * 32 + lo;
        #pragma unroll
        for (int v = 0; v < 8; ++v) {
            float v0 = (*accs0)[v] + bv0;
            float v1 = (*accs1)[v] + bv1;
            if (relu) { v0 = fmaxf(v0, 0.0f); v1 = fmaxf(v1, 0.0f); }
            drow[(size_t)v * ldd]      = v0;
            drow[(size_t)v * ldd + 16] = v1;
        }
    }
}

// ---------------- per-(node,head) attention dot products ----------------
__global__ void attn_reduce_kernel(const float* __restrict__ h,     // [N, 2112]
                                   const float* __restrict__ att_src,
                                   const float* __restrict__ att_dst,
                                   float* __restrict__ a_s, float* __restrict__ a_d) {
    int n    = blockIdx.x;
    int hh   = threadIdx.x >> 5;                // blockDim.x == 224 -> 7 waves
    int lane = threadIdx.x & 31;
    const float* hrow = h + (size_t)n * D_HID_PAD + hh * CH;
    const float* as   = att_src + hh * CH;
    const float* ad   = att_dst + hh * CH;
    float s1 = 0.0f, s2 = 0.0f;
    for (int c = lane; c < CH; c += 32) {
        float v = hrow[c];
        s1 += v * as[c];
        s2 += v * ad[c];
    }
    for (int o = 16; o > 0; o >>= 1) {
        s1 += __shfl_down(s1, o, 32);
        s2 += __shfl_down(s2, o, 32);
    }
    if (lane == 0) {
        a_s[n * HEADS + hh] = s1;
        a_d[n * HEADS + hh] = s2;
    }
}

// ---------------- per-destination softmax + weighted aggregation ----------------
// One block per destination node. Wave 0 builds an ORDER-STABLE edge list via
// ballot compaction (deterministic), then fixed-order softmax and accumulation.
__global__ void gat_aggregate_kernel(const int* __restrict__ src,
                                     const int* __restrict__ dst,
                                     const float* __restrict__ spa,
                                     const float* __restrict__ a_s,
                                     const float* __restrict__ a_d,
                                     const float* __restrict__ h,      // [N, 2112]
                                     const float* __restrict__ bias_gat,
                                     float* __restrict__ out) {       // [N, 2100]
    __shared__ int   s_elist[MAXDEG];
    __shared__ int   s_src[MAXDEG];
    __shared__ float s_ew[MAXDEG * HEADS];
    __shared__ int   s_cnt;

    const int n    = blockIdx.x;
    const int tid  = threadIdx.x;
    const int lane = tid & 31;
    const int wv   = tid >> 5;

    if (wv == 0) {                               // stable compaction of edges with dst==n
        int base = 0;
        for (int e0 = 0; e0 < E_EDGES; e0 += 32) {
            int e = e0 + lane;
            bool match = (dst[e] == n);
            unsigned m = (unsigned)__ballot(match);
            if (match) {
                int ofs = base + __popc(m & ((1u << lane) - 1u));
                if (ofs < MAXDEG) s_elist[ofs] = e;
            }
            base += __popc(m);
        }
        if (lane == 0) s_cnt = (base < MAXDEG) ? base : MAXDEG;
    }
    __syncthreads();
    const int cnt = s_cnt;

    for (int j = tid; j < cnt; j += blockDim.x) s_src[j] = src[s_elist[j]];
    __syncthreads();

    // prefetch source rows of h toward the WGP (global_prefetch_b8);
    // each h row = 2112 floats = 66 x 128B cachelines
    for (int i = tid; i < cnt * 66; i += blockDim.x) {
        int j = i / 66, l = i % 66;
        __builtin_prefetch(h + (size_t)s_src[j] * D_HID_PAD + l * 32, 0, 3);
    }

    // edge logits with leaky-relu
    for (int i = tid; i < cnt * HEADS; i += blockDim.x) {
        int j = i / HEADS, hh = i % HEADS;
        float le = a_s[s_src[j] * HEADS + hh] + a_d[n * HEADS + hh] + spa[s_elist[j]];
        s_ew[j * HEADS + hh] = (le > 0.0f) ? le : NEG_SLOPE * le;
    }
    __syncthreads();

    // per-head softmax over this node's edges (serial -> deterministic; cnt ~ 21)
    if (tid < HEADS) {
        float m = -3.4e38f;
        for (int j = 0; j < cnt; ++j) m = fmaxf(m, s_ew[j * HEADS + tid]);
        float dsum = 0.0f;
        for (int j = 0; j < cnt; ++j) {
            float p = expf(s_ew[j * HEADS + tid] - m);
            s_ew[j * HEADS + tid] = p;
            dsum += p;
        }
        float inv = 1.0f / fmaxf(dsum, 1e-16f);
        for (int j = 0; j < cnt; ++j) s_ew[j * HEADS + tid] *= inv;
    }
    __syncthreads();

    // weighted gather-accumulate over channels (h is L2-resident: 34MB << 192MB L2)
    for (int c = tid; c < D_HID; c += blockDim.x) {
        int hh = c / CH;
        float acc = 0.0f;
        for (int j = 0; j < cnt; ++j)
            acc += s_ew[j * HEADS + hh] * h[(size_t)s_src[j] * D_HID_PAD + c];
        out[(size_t)n * D_HID + c] = acc + bias_gat[c];
    }
}

// ---------------- BatchNorm (training-mode batch stats) ----------------
__global__ void bn_stats_kernel(const float* __restrict__ out,
                                float* __restrict__ mu, float* __restrict__ rstd) {
    __shared__ double ssum[256];
    __shared__ double ssq[256];
    const int col = blockIdx.x;                  // 2100 blocks
    const int tid = threadIdx.x;
    double s = 0.0, q = 0.0;
    for (int r = tid; r < N_NODES; r += 256) {
        double v = (double)out[(size_t)r * D_HID + col];
        s += v; q += v * v;
    }
    ssum[tid] = s; ssq[tid] = q;
    __syncthreads();
    for (int o = 128; o > 0; o >>= 1) {
        if (tid < o) { ssum[tid] += ssum[tid + o]; ssq[tid] += ssq[tid + o]; }
        __syncthreads();
    }
    if (tid == 0) {
        double m   = ssum[0] / (double)N_NODES;
        double var = ssq[0] / (double)N_NODES - m * m;
        mu[col]   = (float)m;
        rstd[col] = (float)(1.0 / sqrt(var + BN_EPS));
    }
}

__global__ void bn_apply_kernel(float* __restrict__ buf,
                                const float* __restrict__ mu,
                                const float* __restrict__ rstd,
                                const float* __restrict__ gamma,
                                const float* __restrict__ beta) {
    size_t i = (size_t)blockIdx.x * blockDim.x + threadIdx.x;
    if (i < (size_t)N_NODES * D_HID) {
        int c = (int)(i % D_HID);
        buf[i] = (buf[i] - mu[c]) * rstd[c] * gamma[c] + beta[c];
    }
}

// ---------------- host-side orchestration ----------------
static inline size_t align256(size_t x) { return (x + 255) & ~(size_t)255; }

extern "C" void kernel_launch(void* const* d_in, const int* in_sizes, int n_in,
                              void* d_out, int out_size, void* d_ws, size_t ws_size,
                              hipStream_t stream) {
    (void)in_sizes; (void)n_in; (void)out_size; (void)ws_size;

    const float* x        = (const float*)d_in[0];
    const float* spa_pos  = (const float*)d_in[1];
    const float* W_gat    = (const float*)d_in[2];
    const float* att_src  = (const float*)d_in[3];
    const float* att_dst  = (const float*)d_in[4];
    const float* bias_gat = (const float*)d_in[5];
    const float* emb_in   = (const float*)d_in[6];
    const float* emb_out  = (const float*)d_in[7];
    const float* gamma    = (const float*)d_in[8];
    const float* beta     = (const float*)d_in[9];
    const float* W3       = (const float*)d_in[10];
    const float* b3       = (const float*)d_in[11];
    const int*   edge     = (const int*)d_in[12];
    const int*   src      = edge;
    const int*   dst      = edge + E_EDGES;
    float*       y        = (float*)d_out;

    // workspace carve-out (~78 MB)
    char* ws = (char*)d_ws;
    size_t off = 0;
    int*   in_deg  = (int*)(ws + off);   off += align256(sizeof(int) * N_NODES);
    int*   out_deg = (int*)(ws + off);   off += align256(sizeof(int) * N_NODES);
    float* xc      = (float*)(ws + off); off += align256(sizeof(float) * (size_t)N_NODES * D_GAT);
    float* WgT     = (float*)(ws + off); off += align256(sizeof(float) * (size_t)D_HID_PAD * D_GAT);
    float* hbuf    = (float*)(ws + off); off += align256(sizeof(float) * (size_t)N_NODES * D_HID_PAD);
    float* a_s     = (float*)(ws + off); off += align256(sizeof(float) * (size_t)N_NODES * HEADS);
    float* a_d     = (float*)(ws + off); off += align256(sizeof(float) * (size_t)N_NODES * HEADS);
    float* normbuf = (float*)(ws + off); off += align256(sizeof(float) * (size_t)N_NODES * D_HID);
    float* mu      = (float*)(ws + off); off += align256(sizeof(float) * D_HID);
    float* rstd    = (float*)(ws + off); off += align256(sizeof(float) * D_HID);
    float* W3T     = (float*)(ws + off); off += align256(sizeof(float) * (size_t)D_OUT * D_HID);

    // 1) degrees
    zero_deg_kernel<<<(N_NODES + 255) / 256, 256, 0, stream>>>(in_deg, out_deg);
    count_deg_kernel<<<(E_EDGES + 255) / 256, 256, 0, stream>>>(src, dst, in_deg, out_deg);

    // 2) concat features
    build_xc_kernel<<<N_NODES, D_GAT, 0, stream>>>(x, emb_in, emb_out, in_deg, out_deg, xc);

    // 3) weight transposes (padded for WMMA tiling)
    transpose_wgat_kernel<<<(D_HID_PAD * D_GAT + 255) / 256, 256, 0, stream>>>(W_gat, WgT);
    transpose_w3_kernel<<<(D_OUT * D_HID + 255) / 256, 256, 0, stream>>>(W3, W3T);

    // 4) GEMM1: h[4096,2112] = xc[4096,272] @ W_gat   (fp32 WMMA, 2x2 blocked)
    {
        int nTilesN = D_HID_PAD / 16;                       // 132 (even)
        int nPairs  = (N_NODES / 32) * (nTilesN / 2);       // 128*66 = 8448 waves
        int blocks  = (nPairs + 7) / 8;                     // 8 waves/block
        gemm_wmma_f32_2x2_kernel<<<blocks, 256, 0, stream>>>(
            xc, WgT, hbuf, nullptr,
            N_NODES, D_GAT, nTilesN, D_GAT, D_GAT, D_HID_PAD, /*relu=*/0);
    }

    // 5) attention coefficients a_s, a_d  (7 waves per node)
    attn_reduce_kernel<<<N_NODES, HEADS * 32, 0, stream>>>(hbuf, att_src, att_dst, a_s, a_d);

    // 6) segment softmax + weighted aggregation + bias (deterministic, no float atomics)
    gat_aggregate_kernel<<<N_NODES, 256, 0, stream>>>(src, dst, spa_pos, a_s, a_d,
                                                      hbuf, bias_gat, normbuf);

    // 7) BatchNorm
    bn_stats_kernel<<<D_HID, 256, 0, stream>>>(normbuf, mu, rstd);
    bn_apply_kernel<<<(int)(((size_t)N_NODES * D_HID + 255) / 256), 256, 0, stream>>>(
        normbuf, mu, rstd, gamma, beta);

    // 8) GEMM2 + bias + ReLU: y[4096,256] = relu(norm @ W3 + b3)  (fp32 WMMA, 2x2 blocked)
    {
        int nTilesN = D_OUT / 16;                           // 16 (even)
        int nPairs  = (N_NODES / 32) * (nTilesN / 2);       // 128*8 = 1024 waves
        int blocks  = (nPairs + 7) / 8;                     // 128 blocks
        gemm_wmma_f32_2x2_kernel<<<blocks, 256, 0, stream>>>(
            normbuf, W3T, y, b3,
            N_NODES, D_HID, nTilesN, D_HID, D_HID, D_OUT, /*relu=*/1);
    }
}